// GeometricPreservationLoss_90460601188802
// MI455X (gfx1250) — compile-verified
//
#include <hip/hip_runtime.h>

typedef __attribute__((ext_vector_type(2))) float v2f;
typedef __attribute__((ext_vector_type(4))) float v4f;
typedef __attribute__((ext_vector_type(8))) float v8f;

#define LOSS_EPS 1e-8f
#define TARGET_RATIO 0.3f
#define MARGIN 0.05f
#define VMAX 16

#if defined(__HIP_DEVICE_COMPILE__) && __has_builtin(__builtin_amdgcn_wmma_f32_16x16x4_f32)
#define HAVE_WMMA_F32X4 1
#else
#define HAVE_WMMA_F32X4 0
#endif

// Stats block layout (floats), at ws + 2*N:
//  [0..16)   counts
//  [16..32)  sum radii (new)
//  [32..48)  sum radii (orig)
//  [48..64)  centered ss (new)
//  [64..80)  centered ss (orig)
//  [80..96)  mean (new)
//  [96..112) mean (orig)
//  [112]     hierarchy sum
//  [113]     coverage sum

__global__ void k_zero(float* __restrict__ st) {
    int i = threadIdx.x;
    if (i < 128) st[i] = 0.0f;
}

__global__ __launch_bounds__(256) void k_norms_wmma(
    const float* __restrict__ zA, const float* __restrict__ zO,
    const int*   __restrict__ vals, const float* __restrict__ tradii,
    float* __restrict__ radii, float* __restrict__ oradii,
    float* __restrict__ st, int N, int D, int V)
{
    __shared__ float s_cnt[VMAX];
    __shared__ float s_sn[VMAX];
    __shared__ float s_so[VMAX];
    __shared__ float s_h;

    const int t = threadIdx.x;
    if (t < VMAX) { s_cnt[t] = 0.0f; s_sn[t] = 0.0f; s_so[t] = 0.0f; }
    if (t == VMAX) s_h = 0.0f;
    __syncthreads();

    const int lane        = t & 31;                 // wave32
    const int waveInBlk   = t >> 5;
    const int wavesPerBlk = blockDim.x >> 5;
    const int gwave       = blockIdx.x * wavesPerBlk + waveInBlk;
    const int nwaves      = gridDim.x * wavesPerBlk;
    const int ntiles      = N >> 4;                 // 16 rows per tile

    for (int tile = gwave; tile < ntiles; tile += nwaves) {
        float dA = 0.0f, dO = 0.0f;   // per-lane squared norm of owned row
        bool  active;
        int   row_local;

#if HAVE_WMMA_F32X4
        // A 16x4 f32 layout: lanes 0-15 hold M=lane, K={k0,k0+1};
        // lanes 16-31 hold M=lane-16, K={k0+2,k0+3}. B 4x16 striping coincides,
        // so one b64 load per lane feeds both operands; diag(A*A^T) = row norms.
        {
            const int rloc = lane & 15;
            const int koff = (lane >> 4) << 1;
            const size_t base = (size_t)(tile * 16 + rloc) * (size_t)D + (size_t)koff;
            const float* pA = zA + base;
            const float* pO = zO + base;
            v8f accA = {}; v8f accO = {};
#pragma unroll 4
            for (int k0 = 0; k0 < D; k0 += 4) {
                v2f a = *(const v2f*)(pA + k0);
                v2f o = *(const v2f*)(pO + k0);
                accA = __builtin_amdgcn_wmma_f32_16x16x4_f32(
                           false, a, false, a, (short)0, accA, false, false);
                accO = __builtin_amdgcn_wmma_f32_16x16x4_f32(
                           false, o, false, o, (short)0, accO, false, false);
            }
            // C/D 16x16 f32 layout: VGPR g, lane l -> (M = g + 8*(l>=16), N = l&15).
            // Diagonal M==N lives in: lanes 0-7 (rows 0-7, element = lane) and
            // lanes 24-31 (rows 8-15, element = lane-24).
#pragma unroll
            for (int i = 0; i < 8; ++i) {
                bool sel = (lane == i) || (lane == i + 24);
                dA = sel ? accA[i] : dA;
                dO = sel ? accO[i] : dO;
            }
            active    = (lane < 8) || (lane >= 24);
            row_local = (lane < 8) ? lane : (lane - 16);
        }
#else
        // Fallback: lanes 0-15 each reduce their own row with vector FMAs.
        {
            active    = (lane < 16);
            row_local = lane;
            if (active) {
                const size_t base = (size_t)(tile * 16 + lane) * (size_t)D;
                const float* pA = zA + base;
                const float* pO = zO + base;
                float a0 = 0.f, a1 = 0.f, o0 = 0.f, o1 = 0.f;
                for (int k = 0; k < D; k += 4) {
                    v4f a = *(const v4f*)(pA + k);
                    v4f o = *(const v4f*)(pO + k);
                    a0 = fmaf(a.x, a.x, fmaf(a.y, a.y, a0));
                    a1 = fmaf(a.z, a.z, fmaf(a.w, a.w, a1));
                    o0 = fmaf(o.x, o.x, fmaf(o.y, o.y, o0));
                    o1 = fmaf(o.z, o.z, fmaf(o.w, o.w, o1));
                }
                dA = a0 + a1; dO = o0 + o1;
            }
        }
#endif
        if (active) {
            int   row = tile * 16 + row_local;
            float r   = sqrtf(dA);
            float ro  = sqrtf(dO);
            radii[row]  = r;
            oradii[row] = ro;
            int v = vals[row];
            v = (v < 0) ? 0 : ((v >= V) ? (V - 1) : v);
            float tr = tradii[v];
            float dh = r - tr;
            atomicAdd(&s_h, dh * dh);
            atomicAdd(&s_cnt[v], 1.0f);
            atomicAdd(&s_sn[v], r);
            atomicAdd(&s_so[v], ro);
        }
    }

    __syncthreads();
    if (t < VMAX && t < V) {
        if (s_cnt[t] != 0.0f) atomicAdd(&st[0  + t], s_cnt[t]);
        if (s_sn[t]  != 0.0f) atomicAdd(&st[16 + t], s_sn[t]);
        if (s_so[t]  != 0.0f) atomicAdd(&st[32 + t], s_so[t]);
    }
    if (t == VMAX) atomicAdd(&st[112], s_h);
}

__global__ __launch_bounds__(256) void k_cov(
    const float* __restrict__ logits, const int* __restrict__ targets,
    float* __restrict__ st, int N)
{
    __shared__ float s;
    if (threadIdx.x == 0) s = 0.0f;
    __syncthreads();
    int i = blockIdx.x * blockDim.x + threadIdx.x;
    if (i < N) {
        float l0 = logits[3 * i + 0];
        float l1 = logits[3 * i + 1];
        float l2 = logits[3 * i + 2];
        float m   = fmaxf(l0, fmaxf(l1, l2));
        float lse = m + logf(expf(l0 - m) + expf(l1 - m) + expf(l2 - m));
        int lab = targets[i] + 1;
        lab = (lab < 0) ? 0 : ((lab > 2) ? 2 : lab);
        float ll = (lab == 0) ? l0 : ((lab == 1) ? l1 : l2);
        atomicAdd(&s, lse - ll);
    }
    __syncthreads();
    if (threadIdx.x == 0) atomicAdd(&st[113], s);
}

__global__ void k_means(float* __restrict__ st, int V) {
    int v = threadIdx.x;
    if (v < V) {
        float d = fmaxf(st[v], 1.0f);
        st[80 + v] = st[16 + v] / d;
        st[96 + v] = st[32 + v] / d;
    }
}

__global__ __launch_bounds__(256) void k_ss(
    const float* __restrict__ radii, const float* __restrict__ oradii,
    const int* __restrict__ vals, float* __restrict__ st, int N, int V)
{
    __shared__ float sn[VMAX];
    __shared__ float so[VMAX];
    int t = threadIdx.x;
    if (t < VMAX) { sn[t] = 0.0f; so[t] = 0.0f; }
    __syncthreads();
    int i = blockIdx.x * blockDim.x + t;
    if (i < N) {
        int v = vals[i];
        v = (v < 0) ? 0 : ((v >= V) ? (V - 1) : v);
        float c  = radii[i]  - st[80 + v];
        float co = oradii[i] - st[96 + v];
        atomicAdd(&sn[v], c * c);
        atomicAdd(&so[v], co * co);
    }
    __syncthreads();
    if (t < VMAX && t < V) {
        if (sn[t] != 0.0f) atomicAdd(&st[48 + t], sn[t]);
        if (so[t] != 0.0f) atomicAdd(&st[64 + t], so[t]);
    }
}

__global__ void k_fin(const float* __restrict__ st, float* __restrict__ out,
                      int N, int V)
{
    if (threadIdx.x != 0 || blockIdx.x != 0) return;
    float h = st[112] / (float)N;
    float c = st[113] / (float)N;

    float rich = 0.0f, npres = 0.0f, sep = 0.0f, prev = 0.0f;
    bool hasPrev = false;
    for (int v = 0; v < V; ++v) {
        float cnt = st[v];
        bool present = cnt > 0.0f;
        bool multi   = cnt > 1.0f;
        float denom  = fmaxf(cnt - 1.0f, 1.0f);
        float vn = st[48 + v] / denom;
        float vo = st[64 + v] / denom;
        if (multi) {
            float ratio = (vn + LOSS_EPS) / (vo + LOSS_EPS);
            float d = ratio - TARGET_RATIO;
            rich += d * d;
        }
        if (present) npres += 1.0f;
        float m = st[80 + v];
        if (present && hasPrev) {
            float viol = m - prev + MARGIN;
            sep += (viol > 0.0f) ? viol : 0.0f;
        }
        if (present) prev = m;
        hasPrev = hasPrev || present;
    }
    rich = rich / fmaxf(npres, 1.0f);
    float total = 10.0f * h + 1.0f * c + 0.5f * rich + 2.0f * sep;
    out[0] = total; out[1] = h; out[2] = c; out[3] = rich; out[4] = sep;
}

extern "C" void kernel_launch(void* const* d_in, const int* in_sizes, int n_in,
                              void* d_out, int out_size, void* d_ws, size_t ws_size,
                              hipStream_t stream)
{
    const float* zA      = (const float*)d_in[0];
    const float* zO      = (const float*)d_in[1];
    const int*   vals    = (const int*)d_in[2];
    const float* tradii  = (const float*)d_in[3];
    const float* logits  = (const float*)d_in[4];
    const int*   targets = (const int*)d_in[5];

    const int N = in_sizes[2];
    const int D = in_sizes[0] / N;
    int V = in_sizes[3];
    if (V > VMAX) V = VMAX;

    float* ws    = (float*)d_ws;
    float* radii = ws;
    float* orad  = ws + (size_t)N;
    float* st    = ws + 2 * (size_t)N;

    k_zero<<<1, 128, 0, stream>>>(st);

    const int ntiles = N >> 4;
    const int wavesPerBlk = 8;            // 256 threads = 8 wave32s
    int blocks = (ntiles + wavesPerBlk - 1) / wavesPerBlk;
    if (blocks > 1024) blocks = 1024;
    if (blocks < 1) blocks = 1;
    k_norms_wmma<<<blocks, 256, 0, stream>>>(zA, zO, vals, tradii,
                                             radii, orad, st, N, D, V);

    k_cov<<<(N + 255) / 256, 256, 0, stream>>>(logits, targets, st, N);
    k_means<<<1, 32, 0, stream>>>(st, V);
    k_ss<<<(N + 255) / 256, 256, 0, stream>>>(radii, orad, vals, st, N, V);
    k_fin<<<1, 1, 0, stream>>>(st, (float*)d_out, N, V);
}